// SSM_Block_Up_24713241821824
// MI455X (gfx1250) — compile-verified
//
#include <hip/hip_runtime.h>
#include <math.h>

// ---------------------------------------------------------------------------
// Problem constants (from setup_inputs): B=2, L=4*32*32=4096, c_in=96, c_out=32
// ssm1: d_model=96, d_inner=192, dt_rank=6, d_state=16
// ssm2: d_model=32, d_inner=64,  dt_rank=2, d_state=16
// ---------------------------------------------------------------------------
#define BATCH 2
#define SEQL  4096
#define BL    (BATCH * SEQL)   // 8192 rows, multiple of 16

typedef float v2f __attribute__((ext_vector_type(2)));
typedef float v8f __attribute__((ext_vector_type(8)));

// ---------------------------------------------------------------------------
// 1) bilinear upsample of patches1 (16x16 -> 32x32) + concat patches2
//    -> seq[b*L + l][c], ld = 96, l = n*1024 + h*32 + w
// ---------------------------------------------------------------------------
__global__ void resize_concat_kernel(const float* __restrict__ p1,
                                     const float* __restrict__ p2,
                                     float* __restrict__ seq) {
  int idx = blockIdx.x * blockDim.x + threadIdx.x;
  if (idx >= BL * 96) return;
  int c   = idx % 96;
  int row = idx / 96;
  int l = row % SEQL;
  int b = row / SEQL;
  int w = l & 31;
  int h = (l >> 5) & 31;
  int n = l >> 10;
  float v;
  if (c < 64) {
    float ph = (float)h * (15.0f / 31.0f);
    int h0 = (int)floorf(ph); float fh = ph - (float)h0;
    int h1 = min(h0 + 1, 15);
    float pw = (float)w * (15.0f / 31.0f);
    int w0 = (int)floorf(pw); float fw = pw - (float)w0;
    int w1 = min(w0 + 1, 15);
    const float* base = p1 + (((size_t)(b * 64 + c) * 4 + n) << 8); // *16*16
    float x00 = base[h0 * 16 + w0];
    float x01 = base[h0 * 16 + w1];
    float x10 = base[h1 * 16 + w0];
    float x11 = base[h1 * 16 + w1];
    float top = x00 + (x01 - x00) * fw;
    float bot = x10 + (x11 - x10) * fw;
    v = top + (bot - top) * fh;
  } else {
    v = p2[(((size_t)(b * 32 + (c - 64)) * 4 + n) << 10) + (h << 5) + w];
  }
  seq[(size_t)row * 96 + c] = v;
}

// ---------------------------------------------------------------------------
// zero-pad weight W[N,K] -> Wp[Np,Kp] so GEMMs need no ragged guards
// ---------------------------------------------------------------------------
__global__ void pad_weight_kernel(const float* __restrict__ in,
                                  float* __restrict__ out,
                                  int N, int K, int Np, int Kp) {
  int idx = blockIdx.x * blockDim.x + threadIdx.x;
  if (idx >= Np * Kp) return;
  int k = idx % Kp;
  int n = idx / Kp;
  out[idx] = (n < N && k < K) ? in[n * K + k] : 0.0f;
}

// ---------------------------------------------------------------------------
// 2) fp32 WMMA GEMM:  Y[BL,N] = X[BL,K] @ W[N,K]^T (+bias) (+softplus)
//    K is a template constant (multiple of 4) -> fully unrolled, branch-free:
//    each lane's A/B fragment is one contiguous 8-byte global_load_b64.
//    A 16x4: lanes 0-15 rows M, K={k0,k0+1}; lanes 16-31 K={k0+2,k0+3}
//    B 4x16: lanes 0-15 cols N, K rows {k0,k0+1}; lanes 16-31 {k0+2,k0+3}
//    C/D   : VGPR r -> row (r + 8*hi), col = lane&15
// ---------------------------------------------------------------------------
template <int K, bool BIAS, bool SOFTPLUS>
__global__ void wmma_gemm_kernel(const float* __restrict__ X, int ldx,
                                 const float* __restrict__ W, int ldw,
                                 const float* __restrict__ bias,
                                 float* __restrict__ Y, int ldy) {
  const int tileM = blockIdx.x * 16;
  const int tileN = blockIdx.y * 16;
  const int lane  = threadIdx.x;      // 0..31, one full wave
  const int hi    = lane >> 4;        // 0|1 -> K sub-pair
  const int lid   = lane & 15;
  const int m = tileM + lid;
  const int n = tileN + lid;

  const v2f* __restrict__ Xp = (const v2f*)(X + (size_t)m * ldx);
  const v2f* __restrict__ Wp = (const v2f*)(W + (size_t)n * ldw);

  v8f acc = {};
#pragma unroll
  for (int k0 = 0; k0 < K; k0 += 4) {
    v2f a = Xp[(k0 >> 1) + hi];   // X[m, k0+2*hi .. +1]
    v2f b = Wp[(k0 >> 1) + hi];   // W[n, k0+2*hi .. +1]
    acc = __builtin_amdgcn_wmma_f32_16x16x4_f32(
        false, a, false, b, (short)0, acc, false, false);
  }

  float bval = 0.0f;
  if (BIAS) bval = bias[n];
#pragma unroll
  for (int r = 0; r < 8; ++r) {
    int row = tileM + r + hi * 8;
    float v = acc[r] + bval;
    if (SOFTPLUS) v = (v > 20.0f) ? v : log1pf(__expf(v));
    Y[(size_t)row * ldy + n] = v;
  }
}

// ---------------------------------------------------------------------------
// 3) depthwise causal conv (k=4, pad (3,0)) + bias + SiLU
// ---------------------------------------------------------------------------
__global__ void conv_silu_kernel(const float* __restrict__ xz, int ldxz,
                                 const float* __restrict__ cw,
                                 const float* __restrict__ cb,
                                 float* __restrict__ xc, int d_inner) {
  int idx = blockIdx.x * blockDim.x + threadIdx.x;
  if (idx >= BL * d_inner) return;
  int d  = idx % d_inner;
  int bl = idx / d_inner;
  int l  = bl % SEQL;
  int b  = bl / SEQL;
  float s = cb[d];
#pragma unroll
  for (int j = 0; j < 4; ++j) {
    int ls = l - 3 + j;
    if (ls >= 0) s += cw[d * 4 + j] * xz[(size_t)(b * SEQL + ls) * ldxz + d];
  }
  s = s / (1.0f + __expf(-s));   // SiLU
  xc[(size_t)bl * d_inner + d] = s;
}

// ---------------------------------------------------------------------------
// 4) fused selective scan (sequential over L, fully fused state update)
//    block = 256 threads: n = tid&15 (d_state=16), dloc = tid>>4 (16 channels)
//    h = exp(delta*A)*h + delta*B*xc ; y = (sum_n h*C + xc*D)*silu(z)
// ---------------------------------------------------------------------------
__global__ void scan_kernel(const float* __restrict__ dbl, int lddbl, int dt_rank,
                            const float* __restrict__ delta,
                            const float* __restrict__ xc,
                            const float* __restrict__ xz, int ldxz,
                            const float* __restrict__ A_log,
                            const float* __restrict__ Dp,
                            float* __restrict__ y, int d_inner) {
  const int t    = threadIdx.x;
  const int n    = t & 15;
  const int dloc = t >> 4;
  const int nblk = d_inner >> 4;
  const int b    = blockIdx.x / nblk;
  const int d    = (blockIdx.x % nblk) * 16 + dloc;

  const float Adn = -__expf(A_log[d * 16 + n]);   // A = -exp(A_log)
  const float Dd  = Dp[d];
  float h = 0.0f;
  const size_t base = (size_t)b * SEQL;

  for (int l = 0; l < SEQL; ++l) {
    const size_t row = base + l;
    if (l + 8 < SEQL) {   // hide L2 latency on the serial chain
      __builtin_prefetch(&delta[(row + 8) * d_inner + d], 0, 1);
      __builtin_prefetch(&dbl[(row + 8) * lddbl + dt_rank + n], 0, 1);
    }
    const float dl  = delta[row * d_inner + d];
    const float xcv = xc[row * d_inner + d];
    const float Bn  = dbl[row * lddbl + dt_rank + n];
    const float Cn  = dbl[row * lddbl + dt_rank + 16 + n];

    const float dA = __expf(dl * Adn);
    h = dA * h + dl * Bn * xcv;

    float p = h * Cn;
    p += __shfl_xor(p, 1, 32);
    p += __shfl_xor(p, 2, 32);
    p += __shfl_xor(p, 4, 32);
    p += __shfl_xor(p, 8, 32);

    if (n == 0) {
      const float zv = xz[row * ldxz + d_inner + d];
      const float sz = zv / (1.0f + __expf(-zv));
      y[row * d_inner + d] = (p + xcv * Dd) * sz;
    }
  }
}

// ---------------------------------------------------------------------------
// 5) layernorm over 32 channels (one lane per channel, full-wave reduction)
//    mode 0: out[row*32+c] = LN(x)*w+b
//    mode 1: d_out[((b*32+c)*L)+l] = LN(x)*w+b + resid[row*32+c]  (transposed)
// ---------------------------------------------------------------------------
__global__ void layernorm_kernel(const float* __restrict__ x, int ldx,
                                 const float* __restrict__ w,
                                 const float* __restrict__ bias,
                                 const float* __restrict__ resid,
                                 float* __restrict__ out, int mode) {
  const int t   = threadIdx.x;
  const int c   = t & 31;
  const int row = blockIdx.x * 8 + (t >> 5);
  if (row >= BL) return;

  float v = x[(size_t)row * ldx + c];
  float s = v;
  s += __shfl_xor(s, 1, 32);
  s += __shfl_xor(s, 2, 32);
  s += __shfl_xor(s, 4, 32);
  s += __shfl_xor(s, 8, 32);
  s += __shfl_xor(s, 16, 32);
  const float mean = s * (1.0f / 32.0f);
  float dfl = v - mean;
  float q = dfl * dfl;
  q += __shfl_xor(q, 1, 32);
  q += __shfl_xor(q, 2, 32);
  q += __shfl_xor(q, 4, 32);
  q += __shfl_xor(q, 8, 32);
  q += __shfl_xor(q, 16, 32);
  const float var = q * (1.0f / 32.0f);
  float r = dfl * rsqrtf(var + 1e-5f) * w[c] + bias[c];

  if (mode == 0) {
    out[(size_t)row * 32 + c] = r;
  } else {
    r += resid[(size_t)row * 32 + c];
    const int l = row % SEQL;
    const int b = row / SEQL;
    out[((size_t)(b * 32 + c) * SEQL) + l] = r;   // (b,L,32) -> (b,32,L)
  }
}

// ---------------------------------------------------------------------------
// Orchestration.  Input flattening order (setup_inputs dict, recursive):
//  0 patches1, 1 patches2,
//  2..10  ssm1: in_proj_w, conv_w, conv_b, x_proj_w, dt_proj_w, dt_proj_b,
//               A_log, D, out_proj_w
// 11..19  ssm2: (same 9)
// 20 ln1_w, 21 ln1_b, 22 ln2_w, 23 ln2_b,
// 24 lin_w, 25 lin_b, 26 lin_ln_w, 27 lin_ln_b
// ---------------------------------------------------------------------------
extern "C" void kernel_launch(void* const* d_in, const int* in_sizes, int n_in,
                              void* d_out, int out_size, void* d_ws, size_t ws_size,
                              hipStream_t stream) {
  const float* patches1 = (const float*)d_in[0];
  const float* patches2 = (const float*)d_in[1];

  const float* s1_in_proj_w  = (const float*)d_in[2];
  const float* s1_conv_w     = (const float*)d_in[3];
  const float* s1_conv_b     = (const float*)d_in[4];
  const float* s1_x_proj_w   = (const float*)d_in[5];
  const float* s1_dt_proj_w  = (const float*)d_in[6];
  const float* s1_dt_proj_b  = (const float*)d_in[7];
  const float* s1_A_log      = (const float*)d_in[8];
  const float* s1_D          = (const float*)d_in[9];
  const float* s1_out_proj_w = (const float*)d_in[10];

  const float* s2_in_proj_w  = (const float*)d_in[11];
  const float* s2_conv_w     = (const float*)d_in[12];
  const float* s2_conv_b     = (const float*)d_in[13];
  const float* s2_x_proj_w   = (const float*)d_in[14];
  const float* s2_dt_proj_w  = (const float*)d_in[15];
  const float* s2_dt_proj_b  = (const float*)d_in[16];
  const float* s2_A_log      = (const float*)d_in[17];
  const float* s2_D          = (const float*)d_in[18];
  const float* s2_out_proj_w = (const float*)d_in[19];

  const float* ln1_w = (const float*)d_in[20];
  const float* ln1_b = (const float*)d_in[21];
  const float* ln2_w = (const float*)d_in[22];
  const float* ln2_b = (const float*)d_in[23];
  const float* lin_w = (const float*)d_in[24];
  const float* lin_b = (const float*)d_in[25];
  const float* lin_ln_w = (const float*)d_in[26];
  const float* lin_ln_b = (const float*)d_in[27];

  // workspace layout (floats), ~38 MB total, fits L2
  float* ws    = (float*)d_ws;
  float* seq   = ws;                        // BL*96
  float* res   = seq   + (size_t)BL * 96;   // BL*32
  float* xzb   = res   + (size_t)BL * 32;   // BL*384 (max)
  float* xcb   = xzb   + (size_t)BL * 384;  // BL*192 (max)
  float* dblb  = xcb   + (size_t)BL * 192;  // BL*48 padded [dt|B|C] — also GEMM temp (ld 32)
  float* dltb  = dblb  + (size_t)BL * 48;   // BL*192 (max)
  float* ybuf  = dltb  + (size_t)BL * 192;  // BL*192 (max)
  float* mid   = ybuf  + (size_t)BL * 192;  // BL*32
  float* px1   = mid   + (size_t)BL * 32;   // 48*192  padded s1_x_proj_w
  float* pdt1  = px1   + 48 * 192;          // 192*8   padded s1_dt_proj_w
  float* px2   = pdt1  + 192 * 8;           // 48*64   padded s2_x_proj_w
  float* pdt2  = px2   + 48 * 64;           // 64*4    padded s2_dt_proj_w
  (void)ws_size; (void)n_in; (void)in_sizes; (void)out_size;

  const dim3 wv(32);
  const int MT = BL / 16;   // 512 M-tiles

  // 0) pad ragged weights (zero rows/cols contribute nothing)
  pad_weight_kernel<<<(48 * 192 + 255) / 256, 256, 0, stream>>>(s1_x_proj_w, px1, 38, 192, 48, 192);
  pad_weight_kernel<<<(192 * 8 + 255) / 256, 256, 0, stream>>>(s1_dt_proj_w, pdt1, 192, 6, 192, 8);
  pad_weight_kernel<<<(48 * 64 + 255) / 256, 256, 0, stream>>>(s2_x_proj_w, px2, 34, 64, 48, 64);
  pad_weight_kernel<<<(64 * 4 + 255) / 256, 256, 0, stream>>>(s2_dt_proj_w, pdt2, 64, 2, 64, 4);

  // 1) build seq
  resize_concat_kernel<<<(BL * 96 + 255) / 256, 256, 0, stream>>>(patches1, patches2, seq);

  // 2) residual branch: LN(seq @ lin_w^T + lin_b)
  wmma_gemm_kernel<96, true, false><<<dim3(MT, 2), wv, 0, stream>>>(seq, 96, lin_w, 96, lin_b, dblb, 32);
  layernorm_kernel<<<BL / 8, 256, 0, stream>>>(dblb, 32, lin_ln_w, lin_ln_b, nullptr, res, 0);

  // ---------------- ssm1: d_model=96, d_inner=192, dt_rank=6 ----------------
  wmma_gemm_kernel<96, false, false><<<dim3(MT, 24), wv, 0, stream>>>(seq, 96, s1_in_proj_w, 96, nullptr, xzb, 384);
  conv_silu_kernel<<<(BL * 192 + 255) / 256, 256, 0, stream>>>(xzb, 384, s1_conv_w, s1_conv_b, xcb, 192);
  wmma_gemm_kernel<192, false, false><<<dim3(MT, 3), wv, 0, stream>>>(xcb, 192, px1, 192, nullptr, dblb, 48);
  wmma_gemm_kernel<8, true, true><<<dim3(MT, 12), wv, 0, stream>>>(dblb, 48, pdt1, 8, s1_dt_proj_b, dltb, 192);
  scan_kernel<<<BATCH * (192 / 16), 256, 0, stream>>>(dblb, 48, 6, dltb, xcb, xzb, 384,
                                                      s1_A_log, s1_D, ybuf, 192);
  wmma_gemm_kernel<192, false, false><<<dim3(MT, 2), wv, 0, stream>>>(ybuf, 192, s1_out_proj_w, 192, nullptr, dblb, 32);
  layernorm_kernel<<<BL / 8, 256, 0, stream>>>(dblb, 32, ln1_w, ln1_b, nullptr, mid, 0);

  // ---------------- ssm2: d_model=32, d_inner=64, dt_rank=2 -----------------
  wmma_gemm_kernel<32, false, false><<<dim3(MT, 8), wv, 0, stream>>>(mid, 32, s2_in_proj_w, 32, nullptr, xzb, 128);
  conv_silu_kernel<<<(BL * 64 + 255) / 256, 256, 0, stream>>>(xzb, 128, s2_conv_w, s2_conv_b, xcb, 64);
  wmma_gemm_kernel<64, false, false><<<dim3(MT, 3), wv, 0, stream>>>(xcb, 64, px2, 64, nullptr, dblb, 48);
  wmma_gemm_kernel<4, true, true><<<dim3(MT, 4), wv, 0, stream>>>(dblb, 48, pdt2, 4, s2_dt_proj_b, dltb, 64);
  scan_kernel<<<BATCH * (64 / 16), 256, 0, stream>>>(dblb, 48, 2, dltb, xcb, xzb, 128,
                                                     s2_A_log, s2_D, ybuf, 64);
  wmma_gemm_kernel<64, false, false><<<dim3(MT, 2), wv, 0, stream>>>(ybuf, 64, s2_out_proj_w, 64, nullptr, dblb, 32);

  // final LN + residual add + transpose to (b, 32, 4, 32, 32)
  layernorm_kernel<<<BL / 8, 256, 0, stream>>>(dblb, 32, ln2_w, ln2_b, res, (float*)d_out, 1);
}